// local_emb_D_1357209665573
// MI455X (gfx1250) — compile-verified
//
#include <hip/hip_runtime.h>
#include <math.h>

typedef __attribute__((ext_vector_type(2))) float v2f;
typedef __attribute__((ext_vector_type(8))) float v8f;

#define HIDDEN 64
#define NEPS 1e-12f

// ---------------------------------------------------------------------------
// Kernel 1: row-wise L2 normalization. One wave32 per row of 64 floats:
// each lane owns a float2 (32 lanes x 8B = 256B row), tree-reduce the sum of
// squares across the wave, scale, and write the normalized row to workspace.
// ---------------------------------------------------------------------------
__global__ void normalize_rows_kernel(const float* __restrict__ emb,
                                      float* __restrict__ out,
                                      int nrows) {
    int gid  = blockIdx.x * blockDim.x + threadIdx.x;
    int row  = gid >> 5;          // one wave (32 lanes) per row
    int lane = gid & 31;
    if (row >= nrows) return;     // wave-uniform branch

    const float2* src = reinterpret_cast<const float2*>(emb + (size_t)row * HIDDEN);
    float2 x = src[lane];
    float s = x.x * x.x + x.y * x.y;
#pragma unroll
    for (int off = 16; off > 0; off >>= 1)
        s += __shfl_xor(s, off, 32);

    float inv = 1.0f / fmaxf(sqrtf(s), NEPS);   // x / max(||x||, eps)
    float2 y; y.x = x.x * inv; y.y = x.y * inv;
    reinterpret_cast<float2*>(out + (size_t)row * HIDDEN)[lane] = y;
}

// ---------------------------------------------------------------------------
// Kernel 2: per-edge weighted dot products via V_WMMA_F32_16X16X4_F32.
// One wave handles 16 edges. A-tile rows = embn[u[e]] (16x64), B-tile cols =
// d * embn[v[e]] (64x16). C = A x B accumulated over K in chunks of 4; the
// diagonal of C holds the 16 per-edge dots.
//
// Per-step operand layout (f32 16x16x4 WMMA, wave32):
//   A vgpr0/1: lanes 0-15 -> K=kk+0,kk+1 of row u[lane]; lanes 16-31 -> K=kk+2,kk+3
//   B vgpr0/1: lanes 0-15 -> K=kk+0,kk+1 of row v[lane]; lanes 16-31 -> K=kk+2,kk+3
// => both A and B are one float2 load per lane at offset kk + 2*(lane>>4).
// C/D layout: lane 0-15 vgpr r -> (M=r, N=lane); lane 16-31 vgpr r -> (M=r+8, N=lane-16)
// => diag e<8 at lane e, vgpr e; diag e>=8 at lane e+16, vgpr e-8.
// ---------------------------------------------------------------------------
__global__ void edge_dot_wmma_kernel(const float* __restrict__ embn,
                                     const float* __restrict__ dvec,
                                     const float* __restrict__ scale,
                                     const int* __restrict__ eidx,
                                     float* __restrict__ out,
                                     int nedges, int ntiles) {
    int wave = (blockIdx.x * blockDim.x + threadIdx.x) >> 5;
    int lane = threadIdx.x & 31;
    if (wave >= ntiles) return;   // wave-uniform: EXEC stays all-ones for WMMA

    int elocal = lane & 15;
    int hi     = lane >> 4;       // 0: K pair (0,1), 1: K pair (2,3)
    int edge   = wave * 16 + elocal;
    int ecl    = edge < nedges ? edge : nedges - 1;   // clamp tail (no divergence)

    int u = eidx[ecl];
    int v = eidx[nedges + ecl];

    const float* Au = embn + (size_t)u * HIDDEN;
    const float* Bv = embn + (size_t)v * HIDDEN;

    v8f c0 = {};
    v8f c1 = {};
#pragma unroll
    for (int kk = 0; kk < HIDDEN; kk += 8) {
        int off0 = kk + 2 * hi;
        int off1 = kk + 4 + 2 * hi;
        v2f a0 = *reinterpret_cast<const v2f*>(Au + off0);
        v2f b0 = *reinterpret_cast<const v2f*>(Bv + off0);
        v2f d0 = *reinterpret_cast<const v2f*>(dvec + off0);
        v2f a1 = *reinterpret_cast<const v2f*>(Au + off1);
        v2f b1 = *reinterpret_cast<const v2f*>(Bv + off1);
        v2f d1 = *reinterpret_cast<const v2f*>(dvec + off1);
        b0 *= d0;
        b1 *= d1;
        // Two independent accumulator chains to hide WMMA RAW latency.
        c0 = __builtin_amdgcn_wmma_f32_16x16x4_f32(false, a0, false, b0,
                                                   (short)0, c0, false, false);
        c1 = __builtin_amdgcn_wmma_f32_16x16x4_f32(false, a1, false, b1,
                                                   (short)0, c1, false, false);
    }
    v8f c = c0 + c1;   // matrix add of partial accumulators

    float sc = scale[0];

    // Diagonal extraction: each producing lane owns exactly one edge result.
    int sel = -1, eo = 0;
    if (lane < 8)        { sel = lane;      eo = lane;      }
    else if (lane >= 24) { sel = lane - 24; eo = lane - 16; }

    if (sel >= 0) {
        float r = 0.0f;
        switch (sel) {
            case 0: r = c[0]; break;
            case 1: r = c[1]; break;
            case 2: r = c[2]; break;
            case 3: r = c[3]; break;
            case 4: r = c[4]; break;
            case 5: r = c[5]; break;
            case 6: r = c[6]; break;
            case 7: r = c[7]; break;
        }
        int eg = wave * 16 + eo;
        if (eg < nedges) out[eg] = r * sc;
    }
}

// ---------------------------------------------------------------------------
// Harness entry point.
//   d_in[0]: emb   [N_NODES * 64] f32
//   d_in[1]: d     [64] f32
//   d_in[2]: scale [1] f32
//   d_in[3]: edge_index [2 * N_EDGES] int
//   d_out  : [N_EDGES] f32
//   d_ws   : >= N_NODES*64*4 bytes for the normalized embedding table
// ---------------------------------------------------------------------------
extern "C" void kernel_launch(void* const* d_in, const int* in_sizes, int n_in,
                              void* d_out, int out_size, void* d_ws, size_t ws_size,
                              hipStream_t stream) {
    const float* emb   = (const float*)d_in[0];
    const float* dvec  = (const float*)d_in[1];
    const float* scale = (const float*)d_in[2];
    const int*   eidx  = (const int*)d_in[3];
    float*       out   = (float*)d_out;

    int nrows  = in_sizes[0] / HIDDEN;
    int nedges = in_sizes[3] / 2;

    float* embn = (float*)d_ws;   // 25.6 MB normalized table (L2-resident)

    // Phase 1: normalize rows. One wave per row, 8 waves per 256-thread block.
    {
        long long total_threads = (long long)nrows * 32;
        int threads = 256;
        int blocks  = (int)((total_threads + threads - 1) / threads);
        normalize_rows_kernel<<<blocks, threads, 0, stream>>>(emb, embn, nrows);
    }

    // Phase 2: 16 edges per wave via WMMA.
    {
        int ntiles = (nedges + 15) / 16;
        long long total_threads = (long long)ntiles * 32;
        int threads = 256;
        int blocks  = (int)((total_threads + threads - 1) / threads);
        edge_dot_wmma_kernel<<<blocks, threads, 0, stream>>>(
            embn, dvec, scale, eidx, out, nedges, ntiles);
    }
}